// fullNN_40973988004624
// MI455X (gfx1250) — compile-verified
//
#include <hip/hip_runtime.h>
#include <math.h>

// ---------------------------------------------------------------------------
// Deep-BSDE forward solver, persistent-kernel formulation for MI455X (gfx1250).
//   B=32768 paths, T=100 steps, per-step MLP 6->12->12->6 with full-batch BN.
//   - 128 WGs x 256 threads (8 wave32 per WG), 1 path per lane, state in regs.
//   - 4 grid-wide BN reductions per step via L2 atomics + arrive counters.
//   - Matmuls via V_WMMA_F32_16X16X4_F32 on zero-padded LDS tiles: all
//     fragment loads unconditional (no exec-mask dances), B-frags hoisted and
//     shared across the two row-tiles, WMMAs interleaved on 2 accumulators.
// ---------------------------------------------------------------------------

#define B_N    32768
#define T_N    100
#define DT_F   0.01f
#define EPS_F  1e-6f
#define ZD     6
#define H_N    12
#define NWG    128
#define TPB    256
#define NWAVE  8
#define NPHASE ((T_N - 1) * 4)   // 396 reduction phases, each with its own slot

typedef __attribute__((ext_vector_type(2))) float v2f;
typedef __attribute__((ext_vector_type(8))) float v8f;

// per-step weight block packed in LDS (zero-padded to WMMA-friendly shapes)
#define OW1P  0     // W1 padded  8x16 (valid 6x12)
#define OW2P  128   // W2 padded 12x16 (valid 12x12)
#define OW3P  320   // W3 padded 12x16 (valid 12x6)
#define OB3P  512   // b3 padded 16    (valid 6)
#define OG0   528
#define OBE0  534
#define OG1   540
#define OBE1  552
#define OG2   564
#define OBE2  576
#define OG3   588
#define OBE3  594
#define WTOT  600

__device__ __forceinline__ float waveSum(float v) {
#pragma unroll
  for (int off = 16; off > 0; off >>= 1) v += __shfl_xor(v, off, 32);
  return v;
}

// Two 16x16 tiles of (A[16xKP] @ Wp[KPx16]) via V_WMMA_F32_16X16X4_F32.
// KP is the zero-padded K (multiple of 4). All LDS loads unconditional:
// pad columns/rows of both A-tiles and Wp are guaranteed zero.
// A-frag: lanes 0-15 hold row m, K={kk,kk+1}; lanes 16-31 hold K={kk+2,kk+3}.
// B-frag: vgpr v, lanes 0-15 -> K=kk+v, lanes 16-31 -> K=kk+v+2, N=lane&15.
template <int KP>
__device__ __forceinline__ void mm_pair(const float* __restrict__ aR0,
                                        const float* __restrict__ aR1,
                                        const float* __restrict__ wp,
                                        int lane, v8f* c0, v8f* c1) {
  const int m  = lane & 15;
  const int hi = lane >> 4;
  v8f acc0 = {0.f, 0.f, 0.f, 0.f, 0.f, 0.f, 0.f, 0.f};
  v8f acc1 = {0.f, 0.f, 0.f, 0.f, 0.f, 0.f, 0.f, 0.f};
#pragma unroll
  for (int kk = 0; kk < KP; kk += 4) {
    const int ka = kk + 2 * hi;
    v2f a0, a1, b;
    a0.x = aR0[m * 17 + ka];
    a0.y = aR0[m * 17 + ka + 1];
    a1.x = aR1[m * 17 + ka];
    a1.y = aR1[m * 17 + ka + 1];
    b.x  = wp[ka * 16 + m];
    b.y  = wp[(ka + 1) * 16 + m];
    acc0 = __builtin_amdgcn_wmma_f32_16x16x4_f32(false, a0, false, b, (short)0,
                                                 acc0, false, false);
    acc1 = __builtin_amdgcn_wmma_f32_16x16x4_f32(false, a1, false, b, (short)0,
                                                 acc1, false, false);
  }
  *c0 = acc0;
  *c1 = acc1;
}

// D-fragment -> row-major LDS tile (row = v + 8*hi, col = lane&15)
__device__ __forceinline__ void store_frag(const v8f d, float* __restrict__ base,
                                           int lane) {
  const int n  = lane & 15;
  const int hi = lane >> 4;
#pragma unroll
  for (int v = 0; v < 8; ++v) base[(v + 8 * hi) * 17 + n] = d[v];
}

// per-channel sum / sum-of-squares of two D-fragments into WG accumulators
__device__ __forceinline__ void frag_reduce(const v8f d0, const v8f d1, int C,
                                            int lane, float* lds_acc) {
  float ps = 0.f, ps2 = 0.f;
#pragma unroll
  for (int v = 0; v < 8; ++v) {
    float a = d0[v], b = d1[v];
    ps  += a + b;
    ps2 += a * a + b * b;
  }
  ps  += __shfl_xor(ps, 16, 32);   // fold lane L and L+16 (same channel)
  ps2 += __shfl_xor(ps2, 16, 32);
  if (lane < C) {
    atomicAdd(&lds_acc[lane], ps);        // ds_add_f32
    atomicAdd(&lds_acc[C + lane], ps2);
  }
}

// Grid-wide BN reduction: WG partials -> L2 atomics -> arrive counter ->
// spin (s_sleep) -> per-channel scale/shift broadcast through LDS.
// scale/shift are zeroed for pad channels so BN-applied pad columns stay 0.
__device__ void bn_finalize(int phase, int C, const float* __restrict__ g,
                            const float* __restrict__ be, float* lds_acc,
                            float* lds_scale, float* lds_shift,
                            float* __restrict__ gacc, int* __restrict__ gcnt,
                            int tid) {
  __syncthreads();  // WG partials complete in lds_acc
  if (tid < 2 * C) {
    atomicAdd(&gacc[phase * 24 + tid], lds_acc[tid]);
    __threadfence();
  }
  __syncthreads();
  if (tid == 0) {
    __hip_atomic_fetch_add(&gcnt[phase], 1, __ATOMIC_RELEASE,
                           __HIP_MEMORY_SCOPE_AGENT);
    while (__hip_atomic_load(&gcnt[phase], __ATOMIC_ACQUIRE,
                             __HIP_MEMORY_SCOPE_AGENT) < NWG)
      __builtin_amdgcn_s_sleep(2);
  }
  __syncthreads();
  if (tid < 24) lds_acc[tid] = 0.f;  // ready for next phase
  if (tid < 16) {
    float sc = 0.f, sh = 0.f;
    if (tid < C) {
      float s  = __hip_atomic_load(&gacc[phase * 24 + tid], __ATOMIC_RELAXED,
                                   __HIP_MEMORY_SCOPE_AGENT);
      float s2 = __hip_atomic_load(&gacc[phase * 24 + C + tid], __ATOMIC_RELAXED,
                                   __HIP_MEMORY_SCOPE_AGENT);
      float mean = s * (1.0f / (float)B_N);
      float var  = fmaxf(s2 * (1.0f / (float)B_N) - mean * mean, 0.f);
      float rs   = rsqrtf(var + EPS_F);
      sc = g[tid] * rs;
      sh = be[tid] - sc * mean;
    }
    lds_scale[tid] = sc;
    lds_shift[tid] = sh;
  }
  __syncthreads();
}

__global__ __launch_bounds__(TPB) void bsde_deep_kernel(
    const float* __restrict__ Wg,  const float* __restrict__ Sg,
    const float* __restrict__ y0i, const float* __restrict__ y1i,
    const float* __restrict__ z0i, const float* __restrict__ z1i,
    const float* __restrict__ W1g, const float* __restrict__ W2g,
    const float* __restrict__ W3g, const float* __restrict__ b3g,
    const float* __restrict__ g0g, const float* __restrict__ be0g,
    const float* __restrict__ g1g, const float* __restrict__ be1g,
    const float* __restrict__ g2g, const float* __restrict__ be2g,
    const float* __restrict__ g3g, const float* __restrict__ be3g,
    float* __restrict__ out, float* __restrict__ gacc, int* __restrict__ gcnt) {
  __shared__ float wlds[WTOT];
  __shared__ float lds_acc[24];
  __shared__ float lds_scale[16];
  __shared__ float lds_shift[16];
  __shared__ float stage[NWAVE][32 * 17];  // per-wave 32x16 staging tile (+pad)

  const int tid  = threadIdx.x;
  const int wv   = tid >> 5;
  const int lane = tid & 31;
  const size_t r = (size_t)(blockIdx.x * NWAVE + wv) * 32 + lane;  // path id

  float* myStage = stage[wv];
  if (tid < 24) lds_acc[tid] = 0.f;
  // zero stage tiles once: guarantees zin pad columns (6..7) read by the
  // first K=8 matmul are zero in step 0; afterwards the h3 store keeps them 0.
  for (int i = tid; i < NWAVE * 32 * 17; i += TPB) (&stage[0][0])[i] = 0.f;

  // initial carry (all in registers for the whole solve)
  float x0 = 100.f, x10 = 1.f, x11 = 1.f;
  float y0 = y0i[0], y10 = y1i[0], y11 = y1i[1];
  float z00 = z0i[0], z01 = z0i[1];
  float z1_00 = z1i[0], z1_01 = z1i[1], z1_10 = z1i[2], z1_11 = z1i[3];

  const float* sRow0 = Sg + r * 2 * T_N;        // S[r,0,:]
  const float* sRow1 = sRow0 + T_N;             // S[r,1,:]
  const float* wRow  = Wg + r * T_N * 2;        // W[r,:,:]

  for (int t = 0; t < T_N - 1; ++t) {
    __syncthreads();
    // cooperative per-step weight/param load into LDS, zero-padded tiles
    for (int i = tid; i < WTOT; i += TPB) {
      float v;
      if (i < OW2P) {                      // W1 pad 8x16 (valid 6x12)
        int rw = i >> 4, cl = i & 15;
        v = (rw < ZD && cl < H_N) ? W1g[t * 72 + rw * H_N + cl] : 0.f;
      } else if (i < OW3P) {               // W2 pad 12x16 (valid 12x12)
        int j = i - OW2P;
        int rw = j >> 4, cl = j & 15;
        v = (cl < H_N) ? W2g[t * 144 + rw * H_N + cl] : 0.f;
      } else if (i < OB3P) {               // W3 pad 12x16 (valid 12x6)
        int j = i - OW3P;
        int rw = j >> 4, cl = j & 15;
        v = (cl < ZD) ? W3g[t * 72 + rw * ZD + cl] : 0.f;
      } else if (i < OG0) {                // b3 pad 16 (valid 6)
        int j = i - OB3P;
        v = (j < ZD) ? b3g[t * 6 + j] : 0.f;
      }
      else if (i < OBE0) v = g0g[t * 6  + (i - OG0)];
      else if (i < OG1)  v = be0g[t * 6  + (i - OBE0)];
      else if (i < OBE1) v = g1g[t * 12  + (i - OG1)];
      else if (i < OG2)  v = be1g[t * 12 + (i - OBE1)];
      else if (i < OBE2) v = g2g[t * 12  + (i - OG2)];
      else if (i < OG3)  v = be2g[t * 12 + (i - OBE2)];
      else if (i < OBE3) v = g3g[t * 6   + (i - OG3)];
      else               v = be3g[t * 6  + (i - OBE3)];
      wlds[i] = v;
    }

    float s0  = sRow0[t];
    float s1  = sRow1[t];
    float w00 = wRow[t * 2 + 0], w01 = wRow[t * 2 + 1];
    float w10 = wRow[t * 2 + 2], w11 = wRow[t * 2 + 3];
    float dw0 = w10 - w00, dw1 = w11 - w01;
    if (t + 1 < T_N - 1) {                    // warm next step's lines
      __builtin_prefetch(&sRow0[t + 1], 0, 1);
      __builtin_prefetch(&sRow1[t + 1], 0, 1);
      __builtin_prefetch(&wRow[(t + 1) * 2 + 2], 0, 1);
    }

    // ---- Euler step + carry updates (pure per-lane VALU) ----
    float aV    = 0.025f * (s0 + s1);
    float adsum = 0.02f * (s0 * x10 + s1 * x11);
    float drift = aV * x0 + adsum;
    float sx1   = x10 + x11;
    float d0    = 0.1f * s0 * x0 + 0.03f * s0 * sx1;
    float d1    = 0.1f * s1 * x0 + 0.03f * s1 * sx1;
    float x0n   = x0 + drift * DT_F + d0 * dw0 + d1 * dw1;

    float f0   = -0.05f * y0 + 0.01f * x0n;
    float y0n  = y0 - f0 * DT_F + z00 * dw0 + z01 * dw1;
    float y1n0 = y10 + 0.05f * y10 * DT_F + z1_00 * dw0 + z1_01 * dw1;
    float y1n1 = y11 + 0.05f * y11 * DT_F + z1_10 * dw0 + z1_11 * dw1;
    float x1n0 = x10 - y1n0 * DT_F;
    float x1n1 = x11 - y1n1 * DT_F;

    float zin[ZD] = {x0n, x1n0, x1n1, y0n, y1n0, y1n1};

    // ---- BN0 over batch ----
#pragma unroll
    for (int c = 0; c < ZD; ++c) {
      float s = waveSum(zin[c]);
      float q = waveSum(zin[c] * zin[c]);
      if (lane == 0) {
        atomicAdd(&lds_acc[c], s);
        atomicAdd(&lds_acc[ZD + c], q);
      }
    }
    bn_finalize(t * 4 + 0, ZD, &wlds[OG0], &wlds[OBE0], lds_acc, lds_scale,
                lds_shift, gacc, gcnt, tid);
#pragma unroll
    for (int c = 0; c < ZD; ++c)
      myStage[lane * 17 + c] = zin[c] * lds_scale[c] + lds_shift[c];
    __syncthreads();

    // ---- layer 1: h0(6) @ W1(6x12), BN, ReLU ----
    v8f a1t0, a1t1;
    mm_pair<8>(myStage, myStage + 16 * 17, &wlds[OW1P], lane, &a1t0, &a1t1);
    frag_reduce(a1t0, a1t1, H_N, lane, lds_acc);
    bn_finalize(t * 4 + 1, H_N, &wlds[OG1], &wlds[OBE1], lds_acc, lds_scale,
                lds_shift, gacc, gcnt, tid);
    {
      int ch = lane & 15;
      float sc = lds_scale[ch], sh = lds_shift[ch];
#pragma unroll
      for (int v = 0; v < 8; ++v) {
        a1t0[v] = fmaxf(a1t0[v] * sc + sh, 0.f);
        a1t1[v] = fmaxf(a1t1[v] * sc + sh, 0.f);
      }
    }
    store_frag(a1t0, myStage, lane);
    store_frag(a1t1, myStage + 16 * 17, lane);
    __syncthreads();

    // ---- layer 2: h1(12) @ W2(12x12), BN, ReLU ----
    v8f a2t0, a2t1;
    mm_pair<12>(myStage, myStage + 16 * 17, &wlds[OW2P], lane, &a2t0, &a2t1);
    frag_reduce(a2t0, a2t1, H_N, lane, lds_acc);
    bn_finalize(t * 4 + 2, H_N, &wlds[OG2], &wlds[OBE2], lds_acc, lds_scale,
                lds_shift, gacc, gcnt, tid);
    {
      int ch = lane & 15;
      float sc = lds_scale[ch], sh = lds_shift[ch];
#pragma unroll
      for (int v = 0; v < 8; ++v) {
        a2t0[v] = fmaxf(a2t0[v] * sc + sh, 0.f);
        a2t1[v] = fmaxf(a2t1[v] * sc + sh, 0.f);
      }
    }
    store_frag(a2t0, myStage, lane);
    store_frag(a2t1, myStage + 16 * 17, lane);
    __syncthreads();

    // ---- layer 3: h2(12) @ W3(12x6) + b3, BN (no ReLU) ----
    v8f a3t0, a3t1;
    mm_pair<12>(myStage, myStage + 16 * 17, &wlds[OW3P], lane, &a3t0, &a3t1);
    {
      int ch = lane & 15;
      float bb = wlds[OB3P + ch];          // zero-padded: no guard needed
#pragma unroll
      for (int v = 0; v < 8; ++v) { a3t0[v] += bb; a3t1[v] += bb; }
    }
    frag_reduce(a3t0, a3t1, ZD, lane, lds_acc);
    bn_finalize(t * 4 + 3, ZD, &wlds[OG3], &wlds[OBE3], lds_acc, lds_scale,
                lds_shift, gacc, gcnt, tid);
    {
      int ch = lane & 15;
      float sc = lds_scale[ch], sh = lds_shift[ch];
#pragma unroll
      for (int v = 0; v < 8; ++v) {
        a3t0[v] = a3t0[v] * sc + sh;       // pad channels stay exactly 0
        a3t1[v] = a3t1[v] * sc + sh;
      }
    }
    store_frag(a3t0, myStage, lane);
    store_frag(a3t1, myStage + 16 * 17, lane);
    __syncthreads();

    // new z carry back to per-lane registers
    z00   = myStage[lane * 17 + 0];
    z01   = myStage[lane * 17 + 1];
    z1_00 = myStage[lane * 17 + 2];
    z1_01 = myStage[lane * 17 + 3];
    z1_10 = myStage[lane * 17 + 4];
    z1_11 = myStage[lane * 17 + 5];

    x0 = x0n; x10 = x1n0; x11 = x1n1;
    y0 = y0n; y10 = y1n0; y11 = y1n1;
  }

  // ---- final partial step (reuses t = T-2 slice of S, last dW) ----
  {
    const int t = T_N - 2;
    float s0  = sRow0[t], s1 = sRow1[t];
    float dw0 = wRow[(T_N - 1) * 2 + 0] - wRow[(T_N - 2) * 2 + 0];
    float dw1 = wRow[(T_N - 1) * 2 + 1] - wRow[(T_N - 2) * 2 + 1];

    float aV    = 0.025f * (s0 + s1);
    float adsum = 0.02f * (s0 * x10 + s1 * x11);
    float drift = aV * x0 + adsum;
    float sx1   = x10 + x11;
    float d0    = 0.1f * s0 * x0 + 0.03f * s0 * sx1;
    float d1    = 0.1f * s1 * x0 + 0.03f * s1 * sx1;
    float x0f   = x0 + drift * DT_F + d0 * dw0 + d1 * dw1;

    float f0   = -0.05f * y0 + 0.01f * x0;   // reference uses pre-final x0 here
    float y0f  = y0 - f0 * DT_F + z00 * dw0 + z01 * dw1;
    float y1f0 = y10 + 0.05f * y10 * DT_F + z1_00 * dw0 + z1_01 * dw1;
    float y1f1 = y11 + 0.05f * y11 * DT_F + z1_10 * dw0 + z1_11 * dw1;
    float x1f0 = x10 - y1f0 * DT_F;
    float x1f1 = x11 - y1f1 * DT_F;

    // outputs concatenated flat: x0f(B) | x1f(B,2) | y0f(B) | y1f(B,2)
    out[r]                   = x0f;
    out[B_N + 2 * r]         = x1f0;
    out[B_N + 2 * r + 1]     = x1f1;
    out[3 * B_N + r]         = y0f;
    out[4 * B_N + 2 * r]     = y1f0;
    out[4 * B_N + 2 * r + 1] = y1f1;
  }
}

extern "C" void kernel_launch(void* const* d_in, const int* in_sizes, int n_in,
                              void* d_out, int out_size, void* d_ws,
                              size_t ws_size, hipStream_t stream) {
  (void)in_sizes; (void)n_in; (void)out_size; (void)ws_size;
  float* gacc = (float*)d_ws;
  int*   gcnt = (int*)((char*)d_ws + (size_t)NPHASE * 24 * sizeof(float));
  size_t clearBytes = (size_t)NPHASE * 24 * sizeof(float) +
                      (size_t)NPHASE * sizeof(int);
  hipMemsetAsync(d_ws, 0, clearBytes, stream);  // fresh slot per phase per call

  bsde_deep_kernel<<<dim3(NWG), dim3(TPB), 0, stream>>>(
      (const float*)d_in[0],  (const float*)d_in[1],  (const float*)d_in[2],
      (const float*)d_in[3],  (const float*)d_in[4],  (const float*)d_in[5],
      (const float*)d_in[6],  (const float*)d_in[7],  (const float*)d_in[8],
      (const float*)d_in[9],  (const float*)d_in[10], (const float*)d_in[11],
      (const float*)d_in[12], (const float*)d_in[13], (const float*)d_in[14],
      (const float*)d_in[15], (const float*)d_in[16], (const float*)d_in[17],
      (float*)d_out, gacc, gcnt);
}